// PeriodicConvolutionPrep_79525614453084
// MI455X (gfx1250) — compile-verified
//
#include <hip/hip_runtime.h>
#include <hip/hip_bf16.h>

typedef float v2f __attribute__((ext_vector_type(2)));
typedef float v8f __attribute__((ext_vector_type(8)));

namespace {
constexpr int kP    = 20000;
constexpr int kDeg  = 16;
constexpr int kCin  = 16;
constexpr int kCout = 16;
constexpr int kTiles = kP / 16;            // 1250 point-tiles of 16
constexpr int kWavesPerBlock = 8;          // 256 threads = 8 wave32
}

// ---------------------------------------------------------------------------
// Stage 1: g[x][p][o] = sum_i features[p][i] * Wk[x][o][i]   (x = 0..2)
// One wave32 per 16-point tile; V_WMMA_F32_16X16X4_F32, K=16 as 4 steps of 4.
// A-frag (16x4 f32): lane L -> (M = L%16, K = 2*(L/16)+v), v in {0,1}
// B-frag (4x16 f32): lane L -> (K = 2*(L/16)+v, N = L%16); B[k][n]=Wk[x][n][k]
// C/D  (16x16 f32): vgpr v -> (M = v + 8*(L/16), N = L%16)
// ---------------------------------------------------------------------------
__global__ __launch_bounds__(256)
void pconv_stage1_wmma(const float* __restrict__ F,   // [P, 16]
                       const float* __restrict__ Wk,  // [3, 16, 16]
                       float* __restrict__ g)         // [3, P, 16] workspace
{
    const int lane = threadIdx.x & 31;
    const int wave = threadIdx.x >> 5;
    const int tile = blockIdx.x * kWavesPerBlock + wave;
    if (tile >= kTiles) return;                 // wave-uniform: EXEC stays all-1s

    const int row = lane & 15;                  // M for A, N for B/C
    const int hi  = lane >> 4;

    // A fragments: one float2 per K-step, 8B-aligned.
    v2f a[4];
    const float* frow = F + (tile * 16 + row) * kCin + 2 * hi;
#pragma unroll
    for (int s = 0; s < 4; ++s)
        a[s] = *(const v2f*)(frow + 4 * s);

#pragma unroll
    for (int x = 0; x < 3; ++x) {
        v8f acc = {};
        const float* wrow = Wk + x * (kCout * kCin) + row * kCin + 2 * hi;
#pragma unroll
        for (int s = 0; s < 4; ++s) {
            v2f b = *(const v2f*)(wrow + 4 * s);
            acc = __builtin_amdgcn_wmma_f32_16x16x4_f32(
                /*neg_a=*/false, a[s], /*neg_b=*/false, b,
                /*c_mod=*/(short)0, acc, /*reuse_a=*/false, /*reuse_b=*/false);
        }
        float* go = g + (size_t)x * (kP * kCout) + tile * (16 * kCout) + row;
#pragma unroll
        for (int v = 0; v < 8; ++v)
            go[(v + 8 * hi) * kCout] = acc[v];  // coalesced 64B per half-wave
    }
}

// ---------------------------------------------------------------------------
// Stage 2: out[p][o] = scale * sum_j sum_x radii[p*16+j][x] * g[x][src[p,j]][o]
// One thread per (point, out-channel); 16 lanes of a point make the g loads
// 64B-coalesced. Everything is L2-resident (g fits in 3.84 MB of 192 MB L2).
// ---------------------------------------------------------------------------
__global__ __launch_bounds__(256)
void pconv_stage2_gather(const float* __restrict__ g,      // [3, P, 16]
                         const float* __restrict__ radii,  // [E, 3]
                         const int*   __restrict__ bs,     // [P, 1+DEG]
                         const int*   __restrict__ n_norm, // scalar
                         float* __restrict__ out)          // [P, 16]
{
    const int gid = blockIdx.x * 256 + threadIdx.x;
    if (gid >= kP * kCout) return;
    const int p = gid >> 4;
    const int o = gid & 15;

    const int*   nbr = bs + p * (1 + kDeg) + 1;
    const float* r   = radii + (size_t)p * kDeg * 3;
    const float* g0  = g;
    const float* g1  = g + (size_t)kP * kCout;
    const float* g2  = g + (size_t)2 * kP * kCout;

    float acc = 0.0f;
#pragma unroll
    for (int j = 0; j < kDeg; ++j) {
        const int   idx  = nbr[j];
        const int   base = idx * kCout + o;
        const float r0 = r[j * 3 + 0];
        const float r1 = r[j * 3 + 1];
        const float r2 = r[j * 3 + 2];
        acc = fmaf(r0, g0[base], acc);
        acc = fmaf(r1, g1[base], acc);
        acc = fmaf(r2, g2[base], acc);
    }
    out[gid] = acc * __frsqrt_rn((float)n_norm[0]);   // 1/sqrt(16) = 0.25
}

extern "C" void kernel_launch(void* const* d_in, const int* in_sizes, int n_in,
                              void* d_out, int out_size, void* d_ws, size_t ws_size,
                              hipStream_t stream)
{
    // setup_inputs() order: features, radii, bs_slice, Wk, n_norm
    const float* features = (const float*)d_in[0];   // [P,16]
    const float* radii    = (const float*)d_in[1];   // [E,3]
    const int*   bs_slice = (const int*)  d_in[2];   // [P,17]
    const float* Wk       = (const float*)d_in[3];   // [3,16,16]
    const int*   n_norm   = (const int*)  d_in[4];   // scalar
    float*       out      = (float*)d_out;           // [P,16]
    float*       g        = (float*)d_ws;            // [3,P,16] = 3.84 MB

    (void)in_sizes; (void)n_in; (void)out_size; (void)ws_size;

    const int blocks1 = (kTiles + kWavesPerBlock - 1) / kWavesPerBlock;   // 157
    pconv_stage1_wmma<<<blocks1, 256, 0, stream>>>(features, Wk, g);

    const int blocks2 = (kP * kCout + 255) / 256;                         // 1250
    pconv_stage2_gather<<<blocks2, 256, 0, stream>>>(g, radii, bs_slice, n_norm, out);
}